// FP_36627481100850
// MI455X (gfx1250) — compile-verified
//
#include <hip/hip_runtime.h>
#include <hip/hip_bf16.h>

// ---------------------------------------------------------------------------
// Shapes (fixed by the reference):
//   real  [32000, 768]  w1 [512, 768]  w2 [4096, 512]  llama [32000, 4096]
//   ids   [4*2048] -> out [8192, 4096]
// Only the gathered 8192 rows are computed (identical per-row math).
// ---------------------------------------------------------------------------

typedef __attribute__((ext_vector_type(2))) float v2f;
typedef __attribute__((ext_vector_type(8))) float v8f;

#define D_REAL  768
#define D_HID   512
#define D_MODEL 4096
#define LDA     772   // 768+4 : stride%64==4 -> conflict-free f32x2 LDS reads
#define LDH     516   // 512+4 : stride%64==4 ; 516*4 = 2064 B (16B aligned)
#define LN_EPS  1e-5f

__device__ __forceinline__ float wave_reduce_sum(float v) {
#pragma unroll
    for (int off = 16; off >= 1; off >>= 1)
        v += __shfl_xor(v, off, 32);
    return v;
}

__device__ __forceinline__ v8f wmma_f32(v2f a, v2f b, v8f c) {
    // V_WMMA_F32_16X16X4_F32 : D = A(16x4,f32) x B(4x16,f32) + C(16x16,f32)
    return __builtin_amdgcn_wmma_f32_16x16x4_f32(
        false, a, false, b, (short)0, c, false, false);
}

// Async global->LDS copy of one 16-byte packet (ASYNCcnt-tracked TDM-lite path).
// lds_byte must be the LDS byte offset; shared-array flat ptr low 32 bits == LDS
// offset per the aperture mapping.
__device__ __forceinline__ void async_ld_b128(unsigned lds_byte, const float* gptr) {
    asm volatile("global_load_async_to_lds_b128 %0, %1, off"
                 :: "v"(lds_byte), "v"(gptr) : "memory");
}
__device__ __forceinline__ void wait_async_all() {
    asm volatile("s_wait_asynccnt 0x0" ::: "memory");
}

// ---------------------------------------------------------------------------
// Stage 1: h[t,:] = ELU(LayerNorm(real[id_t,:] @ w1^T))   -> workspace
// 1 block = 16 tokens, 8 waves; wave w owns hidden cols [64w, 64w+64) (4 tiles)
// ---------------------------------------------------------------------------
__global__ __launch_bounds__(256)
void fp_stage1_wmma(const float* __restrict__ real,
                    const float* __restrict__ w1,
                    const float* __restrict__ ln_w,
                    const float* __restrict__ ln_b,
                    const int*   __restrict__ ids,
                    float*       __restrict__ h_out) {
    __shared__ float lds[16 * LDA];   // A tile; reused as h tile [16*LDH] later
    __shared__ int   sid[16];

    const int tid = threadIdx.x;
    const int tg  = blockIdx.x;                       // token group (16 tokens)
    if (tid < 16) sid[tid] = ids[tg * 16 + tid];
    __syncthreads();

    // Gather 16 rows of `real` into LDS via async-to-LDS b128 (16B per op).
    {
        const unsigned lbase = (unsigned)(size_t)(void*)lds;
        for (int i = tid; i < 16 * (D_REAL / 4); i += 256) {  // 3072 packets
            int m = i / (D_REAL / 4), k = (i - m * (D_REAL / 4)) * 4;
            async_ld_b128(lbase + (unsigned)((m * LDA + k) * 4),
                          real + (long)sid[m] * D_REAL + k);
        }
        wait_async_all();
    }
    __syncthreads();

    const int wave = tid >> 5;
    const int lane = tid & 31;
    const int hn   = lane & 15;          // A row / B,C col within a 16-tile
    const int koff = (lane >> 4) << 1;   // lanes 16-31 hold K+2,K+3
    const int nb   = wave * 64;          // first hidden col of this wave

    v8f acc0 = {}, acc1 = {}, acc2 = {}, acc3 = {};

    for (int kb = 0; kb < D_REAL; kb += 4) {
        v2f a = *(const v2f*)&lds[hn * LDA + kb + koff];
        const float* w1p = w1 + (kb + koff);
        v2f b0 = *(const v2f*)(w1p + (long)(nb      + hn) * D_REAL);
        v2f b1 = *(const v2f*)(w1p + (long)(nb + 16 + hn) * D_REAL);
        v2f b2 = *(const v2f*)(w1p + (long)(nb + 32 + hn) * D_REAL);
        v2f b3 = *(const v2f*)(w1p + (long)(nb + 48 + hn) * D_REAL);
        acc0 = wmma_f32(a, b0, acc0);
        acc1 = wmma_f32(a, b1, acc1);
        acc2 = wmma_f32(a, b2, acc2);
        acc3 = wmma_f32(a, b3, acc3);
    }
    __syncthreads();                      // everyone done reading A tile

    // Spill raw h tiles into LDS for the per-token LayerNorm reduction.
    const int rbase = (lane >> 4) << 3;   // lanes 16-31 hold rows r+8
#pragma unroll
    for (int r = 0; r < 8; ++r) {
        int row = r + rbase;
        lds[row * LDH + nb      + hn] = acc0[r];
        lds[row * LDH + nb + 16 + hn] = acc1[r];
        lds[row * LDH + nb + 32 + hn] = acc2[r];
        lds[row * LDH + nb + 48 + hn] = acc3[r];
    }
    __syncthreads();

    // LayerNorm + ELU: wave w handles tokens 2w and 2w+1.
#pragma unroll
    for (int ti = 0; ti < 2; ++ti) {
        int row = wave * 2 + ti;
        float s = 0.f, s2 = 0.f;
        for (int j = lane; j < D_HID; j += 32) {
            float x = lds[row * LDH + j];
            s += x; s2 += x * x;
        }
        s  = wave_reduce_sum(s);
        s2 = wave_reduce_sum(s2);
        float mu   = s  * (1.f / D_HID);
        float var  = s2 * (1.f / D_HID) - mu * mu;
        float rstd = rsqrtf(var + LN_EPS);
        long  t    = (long)tg * 16 + row;
        for (int j = lane; j < D_HID; j += 32) {
            float x = (lds[row * LDH + j] - mu) * rstd * ln_w[j] + ln_b[j];
            h_out[t * D_HID + j] = (x > 0.f) ? x : expm1f(x);   // ELU(alpha=1)
        }
    }
}

// ---------------------------------------------------------------------------
// Stage 2: out[t,:] = h[t,:] @ w2^T + llama[id_t,:]
// 1 block = 32 tokens x 256 cols; wave owns 2 M-tiles x 2 N-tiles, so every
// B fragment feeds two WMMAs (halves w2 L2 traffic vs 16-token blocks).
// ---------------------------------------------------------------------------
__global__ __launch_bounds__(256)
void fp_stage2_wmma(const float* __restrict__ h_in,
                    const float* __restrict__ w2,
                    const float* __restrict__ llama,
                    const int*   __restrict__ ids,
                    float*       __restrict__ out) {
    __shared__ float hbuf[32 * LDH];     // 66 KB (320 KB/WG budget on CDNA5)
    __shared__ int   sid[32];

    const int tid   = threadIdx.x;
    const int tg    = blockIdx.x;        // token group (32 tokens)
    const int chunk = blockIdx.y;        // 256-wide slice of D_MODEL
    if (tid < 32) sid[tid] = ids[tg * 32 + tid];

    // Async-stage 32 h rows (512 f32 each) into LDS, 16B packets.
    {
        const unsigned lbase = (unsigned)(size_t)(void*)hbuf;
        for (int i = tid; i < 32 * (D_HID / 4); i += 256) {   // 4096 packets
            int m = i >> 7, k = (i & 127) * 4;
            async_ld_b128(lbase + (unsigned)((m * LDH + k) * 4),
                          h_in + ((long)tg * 32 + m) * D_HID + k);
        }
        wait_async_all();
    }
    __syncthreads();

    const int wave = tid >> 5;
    const int lane = tid & 31;
    const int hn   = lane & 15;
    const int koff = (lane >> 4) << 1;
    const int nb   = chunk * 256 + wave * 32;   // 2 N-tiles of 16 cols

    v8f acc00 = {}, acc01 = {}, acc10 = {}, acc11 = {};
    for (int kb = 0; kb < D_HID; kb += 4) {
        v2f a0 = *(const v2f*)&hbuf[(     hn) * LDH + kb + koff];  // tokens 0-15
        v2f a1 = *(const v2f*)&hbuf[(16 + hn) * LDH + kb + koff];  // tokens 16-31
        const float* w2p = w2 + (kb + koff);
        v2f b0 = *(const v2f*)(w2p + (long)(nb      + hn) * D_HID);
        v2f b1 = *(const v2f*)(w2p + (long)(nb + 16 + hn) * D_HID);
        acc00 = wmma_f32(a0, b0, acc00);
        acc10 = wmma_f32(a1, b0, acc10);
        acc01 = wmma_f32(a0, b1, acc01);
        acc11 = wmma_f32(a1, b1, acc11);
    }

    // Epilogue: add gathered llama embedding rows, store f32 output.
    const int rbase = (lane >> 4) << 3;
#pragma unroll
    for (int r = 0; r < 8; ++r) {
        int  row0 = r + rbase;           // M-tile 0 token row
        int  row1 = row0 + 16;           // M-tile 1 token row
        long t0   = (long)tg * 32 + row0;
        long t1   = (long)tg * 32 + row1;
        long l0   = (long)sid[row0] * D_MODEL;
        long l1   = (long)sid[row1] * D_MODEL;
        int  c0   = nb + hn, c1 = nb + 16 + hn;
        out[t0 * D_MODEL + c0] = acc00[r] + llama[l0 + c0];
        out[t0 * D_MODEL + c1] = acc01[r] + llama[l0 + c1];
        out[t1 * D_MODEL + c0] = acc10[r] + llama[l1 + c0];
        out[t1 * D_MODEL + c1] = acc11[r] + llama[l1 + c1];
    }
}

// ---------------------------------------------------------------------------
extern "C" void kernel_launch(void* const* d_in, const int* in_sizes, int n_in,
                              void* d_out, int out_size, void* d_ws, size_t ws_size,
                              hipStream_t stream) {
    const float* real  = (const float*)d_in[0];
    const float* llama = (const float*)d_in[1];
    const float* w1    = (const float*)d_in[2];
    const float* w2    = (const float*)d_in[3];
    const float* ln_w  = (const float*)d_in[4];
    const float* ln_b  = (const float*)d_in[5];
    const int*   ids   = (const int*)d_in[6];   // harness: integer -> int32
    float*       out   = (float*)d_out;
    float*       h_ws  = (float*)d_ws;          // 8192*512 f32 = 16 MB scratch

    const int T = in_sizes[6];                  // B*S = 8192 tokens

    fp_stage1_wmma<<<dim3(T / 16), dim3(256), 0, stream>>>(
        real, w1, ln_w, ln_b, ids, h_ws);
    fp_stage2_wmma<<<dim3(T / 32, D_MODEL / 256), dim3(256), 0, stream>>>(
        h_ws, w2, llama, ids, out);
}